// QueryEncDec_63050119905341
// MI455X (gfx1250) — compile-verified
//
#include <hip/hip_runtime.h>
#include <math.h>

// Problem constants (from reference): SEQ=256, BATCH=2048, LAYERS=128, H=1.
#define T_SEQ    256
#define N_BATCH  2048
#define N_LAYERS 128

// ---------------------------------------------------------------------------
// Fast device transcendentals (CDNA5 has V_EXP_F32 / V_TANH_F32 / V_RCP_F32,
// all TRANS32 ops that co-execute with VALU). Guarded so the host pass and
// older toolchains still compile.
// ---------------------------------------------------------------------------
__device__ __host__ inline float fast_exp2(float x) {
#if __has_builtin(__builtin_amdgcn_exp2f)
  return __builtin_amdgcn_exp2f(x);
#else
  return exp2f(x);
#endif
}

__device__ __host__ inline float fast_rcp(float x) {
#if __has_builtin(__builtin_amdgcn_rcpf)
  return __builtin_amdgcn_rcpf(x);
#else
  return 1.0f / x;
#endif
}

__device__ __host__ inline float fast_sigmoid(float x) {
  // 1 / (1 + exp(-x)) = 1 / (1 + 2^(-x*log2(e)))
  return fast_rcp(1.0f + fast_exp2(-1.4426950408889634f * x));
}

__device__ __host__ inline float fast_tanh(float x) {
#if __has_builtin(__builtin_amdgcn_tanhf)
  return __builtin_amdgcn_tanhf(x);
#elif __has_builtin(__builtin_amdgcn_tanh_f32)
  return __builtin_amdgcn_tanh_f32(x);
#else
  // tanh(x) = (e^{2x}-1)/(e^{2x}+1), e^{2x} = 2^{2x*log2(e)}
  float e = fast_exp2(2.88539008177792681f * x);
  return (e - 1.0f) * fast_rcp(e + 1.0f);
#endif
}

// ---------------------------------------------------------------------------
// TDM availability (device pass only; host pass falls back automatically).
// ---------------------------------------------------------------------------
#if __has_builtin(__builtin_amdgcn_tensor_load_to_lds) && \
    __has_builtin(__builtin_amdgcn_s_wait_tensorcnt)
#define GRU_USE_TDM 1
#endif

typedef unsigned int v4u __attribute__((ext_vector_type(4)));
typedef int          v4i __attribute__((ext_vector_type(4)));
typedef int          v8i __attribute__((ext_vector_type(8)));

// ---------------------------------------------------------------------------
// Wavefront GRU stack kernel.
//   grid  = BATCH blocks (one batch element per block; fully independent)
//   block = LAYERS threads (one GRU layer per thread; 4 wave32 waves)
// Global step s sweeps the (layer, time) anti-diagonal: thread l processes
// time t = s - l, consuming thread (l-1)'s output from the previous step via
// a double-buffered LDS channel (one barrier per step).
// ---------------------------------------------------------------------------
__global__ __launch_bounds__(N_LAYERS) void
gru_stack_wavefront(const float* __restrict__ x,      // [T, B, 1]
                    const float* __restrict__ w_ih,   // [L, 3, 1]
                    const float* __restrict__ w_hh,   // [L, 3, 1]
                    const float* __restrict__ b_ih,   // [L, 3]
                    const float* __restrict__ b_hh,   // [L, 3]
                    float* __restrict__ out)          // [T, B, 1]
{
  __shared__ float xbuf[T_SEQ];           // staged input column x[:, b]
  __shared__ float hbuf[2][N_LAYERS];     // double-buffered layer handoff

  const int l = threadIdx.x;              // layer index
  const int b = blockIdx.x;               // batch element

  // ---- per-layer constants (H == 1 -> all scalars) -----------------------
  // PyTorch gate order: r, z, n. Fold b_ih+b_hh for r,z (n needs them split
  // because gh_n is scaled by r before adding).
  const float wihr = w_ih[l * 3 + 0];
  const float wihz = w_ih[l * 3 + 1];
  const float wihn = w_ih[l * 3 + 2];
  const float whhr = w_hh[l * 3 + 0];
  const float whhz = w_hh[l * 3 + 1];
  const float whhn = w_hh[l * 3 + 2];
  const float br   = b_ih[l * 3 + 0] + b_hh[l * 3 + 0];
  const float bz   = b_ih[l * 3 + 1] + b_hh[l * 3 + 1];
  const float bihn = b_ih[l * 3 + 2];
  const float bhhn = b_hh[l * 3 + 2];

  // ---- stage x[:, b] (stride N_BATCH floats) into LDS --------------------
#ifdef GRU_USE_TDM
  if (threadIdx.x < 32) {  // one full wave issues the TDM op (EXEC all-ones)
    // D# group 0: count=1 | lds_addr | global_addr(57b) | type=2
    const unsigned lds_off =
        (unsigned)(unsigned long long)(const void*)&xbuf[0];  // low 32 bits of
                                                              // flat LDS addr
                                                              // = WG-relative
                                                              // byte offset
    const unsigned long long ga = (unsigned long long)(const void*)(x + b);
    v4u g0;
    g0.x = 1u;                                     // count = 1 (valid D#)
    g0.y = lds_off;                                // lds_addr
    g0.z = (unsigned)(ga & 0xFFFFFFFFull);         // global_addr[31:0]
    g0.w = (unsigned)((ga >> 32) & 0x1FFFFFFull)   // global_addr[56:32]
           | 0x80000000u;                          // type = 2 ("image")

    // D# group 1: 2D tile, 1 x T_SEQ elements of 4B, row stride = N_BATCH.
    v8i g1;
    g1[0] = (int)(2u << 16);                  // data_size = 2 -> 4 bytes
    g1[1] = (int)(1u << 16);                  // tensor_dim0[15:0]  = 1
    g1[2] = (int)((unsigned)T_SEQ << 16);     // tensor_dim1[15:0]  = 256
    g1[3] = (int)(1u << 16);                  // tile_dim0          = 1
    g1[4] = (int)(unsigned)T_SEQ;             // tile_dim1 = 256, tile_dim2 = 0
    g1[5] = (int)N_BATCH;                     // tensor_dim0_stride = 2048
    g1[6] = 0;                                // stride hi / dim1_stride lo
    g1[7] = 0;                                // dim1_stride hi (unused, 2D)

    const v4i gz4 = {0, 0, 0, 0};             // groups 2/3 unused (<=2D)
#if defined(__clang_major__) && __clang_major__ >= 23
    const v8i gz8 = {0, 0, 0, 0, 0, 0, 0, 0};
    __builtin_amdgcn_tensor_load_to_lds(g0, g1, gz4, gz4, gz8, 0);
#else
    __builtin_amdgcn_tensor_load_to_lds(g0, g1, gz4, gz4, 0);
#endif
    __builtin_amdgcn_s_wait_tensorcnt(0);
  }
#else
  for (int i = threadIdx.x; i < T_SEQ; i += N_LAYERS) {
    xbuf[i] = x[(size_t)i * N_BATCH + b];
  }
#endif

  hbuf[0][l] = 0.0f;
  hbuf[1][l] = 0.0f;
  __syncthreads();

  // ---- wavefront over s = l + t ------------------------------------------
  float h = 0.0f;
  const int src = (l == 0) ? 0 : (l - 1);   // clamped neighbor index

  for (int s = 0; s < T_SEQ + N_LAYERS - 1; ++s) {
    const int t = s - l;
    if (t >= 0 && t < T_SEQ) {
      const float prev = hbuf[s & 1][src];                  // layer l-1 @ t
      const float in   = (l == 0) ? xbuf[t] : prev;

      const float gr  = fmaf(in, wihr, fmaf(h, whhr, br));  // r pre-act
      const float gz  = fmaf(in, wihz, fmaf(h, whhz, bz));  // z pre-act
      const float ghn = fmaf(h, whhn, bhhn);                // hidden n part
      const float r   = fast_sigmoid(gr);
      const float z   = fast_sigmoid(gz);
      const float n   = fast_tanh(fmaf(r, ghn, fmaf(in, wihn, bihn)));
      h = fmaf(z, h - n, n);                 // (1-z)*n + z*h

      if (l == N_LAYERS - 1) {
        out[(size_t)t * N_BATCH + b] = h;    // final-layer sequence output
      }
    }
    hbuf[(s + 1) & 1][l] = h;                // publish for layer l+1
    __syncthreads();                          // split barrier on gfx1250
  }
}

// ---------------------------------------------------------------------------
// Harness entry point.
//   d_in order: x [T,B,1], w_ih [L,3,1], w_hh [L,3,1], b_ih [L,3], b_hh [L,3]
//   d_out: [T,B,1] float32
// ---------------------------------------------------------------------------
extern "C" void kernel_launch(void* const* d_in, const int* in_sizes, int n_in,
                              void* d_out, int out_size, void* d_ws,
                              size_t ws_size, hipStream_t stream) {
  (void)in_sizes; (void)n_in; (void)out_size; (void)d_ws; (void)ws_size;

  const float* x    = (const float*)d_in[0];
  const float* w_ih = (const float*)d_in[1];
  const float* w_hh = (const float*)d_in[2];
  const float* b_ih = (const float*)d_in[3];
  const float* b_hh = (const float*)d_in[4];
  float*       out  = (float*)d_out;

  dim3 grid(N_BATCH);     // one block per batch element
  dim3 block(N_LAYERS);   // one thread per layer (4 wave32 waves)
  gru_stack_wavefront<<<grid, block, 0, stream>>>(x, w_ih, w_hh, b_ih, b_hh,
                                                  out);
}